// AdaptiveSpectralBlock_56358560858760
// MI455X (gfx1250) — compile-verified
//
#include <hip/hip_runtime.h>
#include <hip/hip_bf16.h>
#include <math.h>

// Problem constants (match reference)
#define BB 2
#define CC 512
#define DD 1024
#define KK 64
#define FBINS 513
#define LN_EPS 1e-5f

typedef __attribute__((ext_vector_type(2))) float v2f;
typedef __attribute__((ext_vector_type(8))) float v8f;

// ---------------------------------------------------------------------------
// Kernel 1: build M[d, 0:64]=Re, M[d,64:128]=Im  where
//   M[d,k] = sum_f exp(-2*pi*i*f*d/D) * P[f,k]
// Rotation recurrence, re-synced via sincosf every 64 steps (exact phase via
// (d*f) mod 1024 to keep range reduction benign).
// ---------------------------------------------------------------------------
__global__ void build_M_kernel(const float* __restrict__ P, float* __restrict__ Mcat) {
    int idx = blockIdx.x * blockDim.x + threadIdx.x;
    if (idx >= DD * KK) return;
    int d = idx >> 6;
    int k = idx & 63;
    const float theta = -6.283185307179586f * (float)d * (1.0f / 1024.0f);
    float cs, sn;
    sincosf(theta, &sn, &cs);           // e^{i*theta} step
    float wr = 1.0f, wi = 0.0f;         // e^{i*theta*f}
    float sr = 0.0f, si = 0.0f;
    for (int f = 0; f < FBINS; ++f) {
        if ((f & 63) == 0) {            // resync to kill recurrence drift
            int m = (d * f) & 1023;
            float ang = -6.283185307179586f * (float)m * (1.0f / 1024.0f);
            sincosf(ang, &wi, &wr);
        }
        float p = P[f * KK + k];
        sr = fmaf(wr, p, sr);
        si = fmaf(wi, p, si);
        float nwr = wr * cs - wi * sn;  // rotate by e^{i*theta}
        wi = fmaf(wr, sn, wi * cs);
        wr = nwr;
    }
    Mcat[d * (2 * KK) + k]      = sr;
    Mcat[d * (2 * KK) + KK + k] = si;
}

// ---------------------------------------------------------------------------
// Generic fp32 WMMA GEMM: C[M,N] = A[M,Kd] @ B[Kd,N]  (+ optional addend)
// Each wave owns a 16 x (16*NT) strip. Double-buffered fragment loads: the
// next k-step's A + NT B fragments are fetched into fresh registers before
// the current step's WMMAs issue, so global loads overlap the XDL pipe.
// A layout (16x4): lane m = lane&15, half = lane>>4 -> a = {A[m][kk+2h], A[m][kk+2h+1]}
// B layout (4x16): b = {B[kk+2h][n], B[kk+2h+1][n]}
// C layout (16x16): vgpr i holds rows i (lanes 0-15) and i+8 (lanes 16-31)
// ---------------------------------------------------------------------------
template <int NT>
__global__ void wmma_gemm_f32(const float* __restrict__ A, int lda,
                              const float* __restrict__ B, int ldb,
                              float* __restrict__ C, int ldc,
                              const float* __restrict__ addend, int ldadd,
                              int M, int N, int Kd) {
    int wave    = (blockIdx.x * blockDim.x + threadIdx.x) >> 5;
    int ngroups = N / (16 * NT);
    int total   = (M >> 4) * ngroups;
    if (wave >= total) return;                 // wave-uniform: EXEC stays full
    int mt = wave / ngroups;
    int ng = wave - mt * ngroups;
    int row0 = mt << 4;
    int col0 = ng * (16 * NT);

    int lane = threadIdx.x & 31;
    int m    = lane & 15;
    int half = lane >> 4;                      // 0: K pair {0,1}, 1: K pair {2,3}

    v8f acc[NT] = {};
    const float* aptr = A + (size_t)(row0 + m) * lda + 2 * half;
    const float* bptr = B + (size_t)(2 * half) * ldb + col0 + m;

    // ---- prologue: load fragments for k-step 0 ----
    v2f a_cur;
    v2f b_cur[NT];
    a_cur.x = aptr[0];
    a_cur.y = aptr[1];
    #pragma unroll
    for (int j = 0; j < NT; ++j) {
        b_cur[j].x = bptr[j * 16];
        b_cur[j].y = bptr[(size_t)ldb + j * 16];
    }

    int steps = Kd >> 2;
    #pragma unroll 2
    for (int it = 0; it < steps - 1; ++it) {
        aptr += 4;
        bptr += (size_t)4 * ldb;
        // prefetch next k-step into a fresh buffer (overlaps WMMA below)
        v2f a_nxt;
        v2f b_nxt[NT];
        a_nxt.x = aptr[0];
        a_nxt.y = aptr[1];
        #pragma unroll
        for (int j = 0; j < NT; ++j) {
            b_nxt[j].x = bptr[j * 16];
            b_nxt[j].y = bptr[(size_t)ldb + j * 16];
        }
        // consume current fragments
        #pragma unroll
        for (int j = 0; j < NT; ++j)
            acc[j] = __builtin_amdgcn_wmma_f32_16x16x4_f32(
                false, a_cur, false, b_cur[j], (short)0, acc[j], false, false);
        a_cur = a_nxt;
        #pragma unroll
        for (int j = 0; j < NT; ++j) b_cur[j] = b_nxt[j];
    }
    // ---- epilogue k-step ----
    #pragma unroll
    for (int j = 0; j < NT; ++j)
        acc[j] = __builtin_amdgcn_wmma_f32_16x16x4_f32(
            false, a_cur, false, b_cur[j], (short)0, acc[j], false, false);

    int rbase = row0 + (half << 3);
    #pragma unroll
    for (int j = 0; j < NT; ++j) {
        int n = col0 + j * 16 + m;
        if (addend) {
            #pragma unroll
            for (int i = 0; i < 8; ++i)
                C[(size_t)(rbase + i) * ldc + n] =
                    acc[j][i] + addend[(size_t)(rbase + i) * ldadd + n];
        } else {
            #pragma unroll
            for (int i = 0; i < 8; ++i)
                C[(size_t)(rbase + i) * ldc + n] = acc[j][i];
        }
    }
}

// ---------------------------------------------------------------------------
// Kernel 3: per (b,c) row -> filtered real spectrum fr[k], exact-GELU MLP
// score, softmax, emit WK[row,k] = fr[k]*softmax_k(score).
// One block (8 waves) per row; each wave owns 8 k-bins, lanes read float4 of
// E1; b1/w2 staged in LDS once per block. VALU hot spot: 67M erf evals.
// ---------------------------------------------------------------------------
__global__ void score_kernel(const float* __restrict__ S,      // (B*C, 2K) re|im
                             const float* __restrict__ gr, const float* __restrict__ gi,
                             const float* __restrict__ lr, const float* __restrict__ li,
                             const float* __restrict__ E1,     // (K, D) = femb@w1
                             const float* __restrict__ b1,
                             const float* __restrict__ w2,
                             const float* __restrict__ b2v,
                             const float* __restrict__ thr_p,
                             float* __restrict__ WK) {         // (B*C, K)
    __shared__ float frs[KK];
    __shared__ float sc[KK];
    __shared__ float red[2];
    __shared__ float b1s[DD];
    __shared__ float w2s[DD];
    int row = blockIdx.x;
    int c   = row & (CC - 1);
    int tid = threadIdx.x;
    float thr = thr_p[0];

    {   // stage b1 / w2 into LDS (256 threads * float4 = 1024 floats each)
        int d = tid * 4;
        *(float4*)(b1s + d) = *(const float4*)(b1 + d);
        *(float4*)(w2s + d) = *(const float4*)(w2 + d);
    }
    if (tid < KK) {
        float sr = S[row * (2 * KK) + tid];
        float si = S[row * (2 * KK) + KK + tid];
        float mask = (sr * sr + si * si) > thr ? 1.0f : 0.0f;
        int gidx = c * KK + tid;
        float fre = sr * gr[gidx] - si * gi[gidx];
        fre += mask * (sr * lr[gidx] - si * li[gidx]);
        frs[tid] = fre;
    }
    __syncthreads();

    int wave = tid >> 5, lane = tid & 31;
    const float inv_sqrt2 = 0.7071067811865476f;
    for (int k = wave; k < KK; k += 8) {
        float f = frs[k];
        const float* e = E1 + (size_t)k * DD;
        float acc = 0.0f;
        for (int d = lane * 4; d < DD; d += 128) {
            float4 ev = *(const float4*)(e + d);
            float4 bv = *(const float4*)(b1s + d);
            float4 wv = *(const float4*)(w2s + d);
            float t0 = fmaf(f, ev.x, bv.x);
            float t1 = fmaf(f, ev.y, bv.y);
            float t2 = fmaf(f, ev.z, bv.z);
            float t3 = fmaf(f, ev.w, bv.w);
            float g0 = 0.5f * t0 * (1.0f + erff(t0 * inv_sqrt2));
            float g1 = 0.5f * t1 * (1.0f + erff(t1 * inv_sqrt2));
            float g2 = 0.5f * t2 * (1.0f + erff(t2 * inv_sqrt2));
            float g3 = 0.5f * t3 * (1.0f + erff(t3 * inv_sqrt2));
            acc = fmaf(g0, wv.x, acc);
            acc = fmaf(g1, wv.y, acc);
            acc = fmaf(g2, wv.z, acc);
            acc = fmaf(g3, wv.w, acc);
        }
        #pragma unroll
        for (int off = 16; off; off >>= 1) acc += __shfl_xor(acc, off, 32);
        if (lane == 0) sc[k] = acc + b2v[0];
    }
    __syncthreads();

    if (tid == 0) {
        float mx = -INFINITY;
        for (int k = 0; k < KK; ++k) mx = fmaxf(mx, sc[k]);
        float s = 0.0f;
        for (int k = 0; k < KK; ++k) s += expf(sc[k] - mx);
        red[0] = mx; red[1] = 1.0f / s;
    }
    __syncthreads();

    if (tid < KK) {
        float w = expf(sc[tid] - red[0]) * red[1];
        WK[row * KK + tid] = frs[tid] * w;
    }
}

// ---------------------------------------------------------------------------
// Kernel 5: in-place LayerNorm over last dim (biased var, eps=1e-5)
// One block per row; each thread owns exactly one float4 (256*4 = 1024).
// ---------------------------------------------------------------------------
__global__ void ln_kernel(float* __restrict__ X,
                          const float* __restrict__ gamma,
                          const float* __restrict__ beta) {
    __shared__ float ra[8], rb[8];
    int row = blockIdx.x;
    int tid = threadIdx.x;
    float* xr = X + (size_t)row * DD;
    int d = tid * 4;
    float4 v = *(const float4*)(xr + d);
    float s  = v.x + v.y + v.z + v.w;
    float s2 = fmaf(v.x, v.x, fmaf(v.y, v.y, fmaf(v.z, v.z, v.w * v.w)));
    #pragma unroll
    for (int off = 16; off; off >>= 1) {
        s  += __shfl_xor(s,  off, 32);
        s2 += __shfl_xor(s2, off, 32);
    }
    int wave = tid >> 5, lane = tid & 31;
    if (lane == 0) { ra[wave] = s; rb[wave] = s2; }
    __syncthreads();
    if (tid == 0) {
        float ts = 0.0f, ts2 = 0.0f;
        for (int i = 0; i < 8; ++i) { ts += ra[i]; ts2 += rb[i]; }
        float mean = ts * (1.0f / DD);
        float var  = ts2 * (1.0f / DD) - mean * mean;
        ra[0] = mean;
        rb[0] = rsqrtf(var + LN_EPS);
    }
    __syncthreads();
    float mean = ra[0], rstd = rb[0];
    float4 g = *(const float4*)(gamma + d);
    float4 bt = *(const float4*)(beta + d);
    float4 o;
    o.x = (v.x - mean) * rstd * g.x + bt.x;
    o.y = (v.y - mean) * rstd * g.y + bt.y;
    o.z = (v.z - mean) * rstd * g.z + bt.z;
    o.w = (v.w - mean) * rstd * g.w + bt.w;
    *(float4*)(xr + d) = o;
}

// ---------------------------------------------------------------------------
// Host launcher
// ---------------------------------------------------------------------------
template <int NT>
static inline void launch_gemm(const float* A, int lda, const float* B, int ldb,
                               float* C, int ldc, const float* addend, int ldadd,
                               int M, int N, int Kd, hipStream_t stream) {
    int waves   = (M / 16) * (N / (16 * NT));
    int threads = waves * 32;
    int block   = 256;
    int grid    = (threads + block - 1) / block;
    wmma_gemm_f32<NT><<<grid, block, 0, stream>>>(A, lda, B, ldb, C, ldc,
                                                  addend, ldadd, M, N, Kd);
}

extern "C" void kernel_launch(void* const* d_in, const int* in_sizes, int n_in,
                              void* d_out, int out_size, void* d_ws, size_t ws_size,
                              hipStream_t stream) {
    const float* tokens = (const float*)d_in[0];   // (B,C,D)
    const float* thr    = (const float*)d_in[1];   // scalar
    const float* proj   = (const float*)d_in[2];   // (FB,K)
    const float* gr     = (const float*)d_in[3];   // (C,K)
    const float* gi     = (const float*)d_in[4];
    const float* lr     = (const float*)d_in[5];
    const float* li     = (const float*)d_in[6];
    const float* femb   = (const float*)d_in[7];   // (K,D)
    const float* w1     = (const float*)d_in[8];   // (D,D)
    const float* b1     = (const float*)d_in[9];   // (D,)
    const float* w2     = (const float*)d_in[10];  // (D,1)
    const float* b2     = (const float*)d_in[11];  // (1,)
    const float* gamma  = (const float*)d_in[12];  // (D,)
    const float* beta   = (const float*)d_in[13];  // (D,)

    const int rows = BB * CC;                      // 1024
    float* ws   = (float*)d_ws;
    float* Mcat = ws;                              // D x 2K     = 131072 f
    float* S    = Mcat + (size_t)DD * 2 * KK;      // rows x 2K  = 131072 f
    float* E1   = S    + (size_t)rows * 2 * KK;    // K x D      =  65536 f
    float* WK   = E1   + (size_t)KK * DD;          // rows x K   =  65536 f
    float* X    = (float*)d_out;                   // rows x D (in-place LN)

    // 1) DFT-fold matrix M (cols 0..63 = Re, 64..127 = Im)
    build_M_kernel<<<(DD * KK) / 256, 256, 0, stream>>>(proj, Mcat);

    // 2) E1 = femb @ w1                     (64 x 1024 x 1024, WMMA f32)
    launch_gemm<4>(femb, DD, w1, DD, E1, DD, nullptr, 0, KK, DD, DD, stream);

    // 3) S = tokens @ Mcat  (rfft + projection fused)   (1024 x 1024 x 128)
    launch_gemm<2>(tokens, DD, Mcat, 2 * KK, S, 2 * KK, nullptr, 0, rows, 2 * KK, DD, stream);

    // 4) filter -> exact-GELU MLP score -> softmax -> WK = fr * weights
    score_kernel<<<rows, 256, 0, stream>>>(S, gr, gi, lr, li, E1, b1, w2, b2, thr, WK);

    // 5) X = WK @ femb + tokens             (1024 x 64 x 1024, WMMA f32)
    launch_gemm<4>(WK, KK, femb, DD, X, DD, tokens, DD, rows, DD, KK, stream);

    // 6) in-place LayerNorm
    ln_kernel<<<rows, 256, 0, stream>>>(X, gamma, beta);
}